// ReflectionalSwitchFunction_82806969467114
// MI455X (gfx1250) — compile-verified
//
#include <hip/hip_runtime.h>

// RSWAF basis: out[..., g] = 1 - tanh((x - grid[g]) * inv_den)^2  == sech^2
// x: [16,64,128,128] f32 (16,777,216 elems), grid: [8] f32, inv_den: scalar f32
// out: 134,217,728 f32 (512 MB) -> store-bandwidth-bound; floor ~25us @ 23.3TB/s.

typedef __attribute__((ext_vector_type(4))) float v4f;

// 1 - tanh(y)^2 = 4u / (1+u)^2 with u = exp(-2|y|).
// u in (0,1]; underflows to 0 for large |y| -> result cleanly -> 0 (no inf/inf).
// Uses v_exp_f32 + v_rcp_f32 hardware transcendentals.
__device__ __forceinline__ float sech2(float y) {
    const float NEG_2_LOG2E = -2.8853900817779268f; // -2 * log2(e)
    float a  = __builtin_fabsf(y);
    float u  = __builtin_amdgcn_exp2f(a * NEG_2_LOG2E); // v_exp_f32
    float s  = 1.0f + u;
    float rc = __builtin_amdgcn_rcpf(s * s);            // v_rcp_f32
    return (4.0f * u) * rc;
}

// One thread produces 4 consecutive outputs (16 B) -> dense global_store_b128.
// vec index t: x element = t>>1 (8 outputs per x), grid half = (t&1)*4.
__global__ __launch_bounds__(256) void rswaf_vec4_kernel(
    const float* __restrict__ x,
    const float* __restrict__ grid,
    const float* __restrict__ inv_den,
    float* __restrict__ out,
    int nvec)
{
    int t = blockIdx.x * blockDim.x + threadIdx.x;
    if (t >= nvec) return;

    const float invd = inv_den[0];                        // uniform -> s_load
    const float xv   = x[t >> 1];                         // cached, 1/8 of traffic
    const v4f   g    = *(const v4f*)(grid + ((t & 1) << 2)); // 16B of the 32B grid

    v4f r;
#pragma unroll
    for (int i = 0; i < 4; ++i) {
        r[i] = sech2((xv - g[i]) * invd);
    }
    // Output (512 MB) > L2 (192 MB) and never re-read: stream past L2.
    __builtin_nontemporal_store(r, (v4f*)out + t);        // global_store_b128 (NT)
}

// Tail safety for out_size not divisible by 4 (not hit with reference shapes).
__global__ __launch_bounds__(64) void rswaf_tail_kernel(
    const float* __restrict__ x,
    const float* __restrict__ grid,
    const float* __restrict__ inv_den,
    float* __restrict__ out,
    int start, int total)
{
    int idx = start + (int)threadIdx.x + (int)blockIdx.x * (int)blockDim.x;
    if (idx >= total) return;
    const float invd = inv_den[0];
    float xv = x[idx >> 3];
    float gv = grid[idx & 7];
    out[idx] = sech2((xv - gv) * invd);
}

extern "C" void kernel_launch(void* const* d_in, const int* in_sizes, int n_in,
                              void* d_out, int out_size, void* d_ws, size_t ws_size,
                              hipStream_t stream) {
    const float* x    = (const float*)d_in[0];
    const float* grid = (const float*)d_in[1];
    const float* invd = (const float*)d_in[2];
    float* out        = (float*)d_out;

    int nvec = out_size >> 2;                 // float4 chunks
    if (nvec > 0) {
        const int threads = 256;
        int blocks = (nvec + threads - 1) / threads;
        rswaf_vec4_kernel<<<blocks, threads, 0, stream>>>(x, grid, invd, out, nvec);
    }
    int done = nvec << 2;
    if (done < out_size) {
        int rem = out_size - done;
        rswaf_tail_kernel<<<(rem + 63) / 64, 64, 0, stream>>>(x, grid, invd, out, done, out_size);
    }
}